// Group_81140522156683
// MI455X (gfx1250) — compile-verified
//
#include <hip/hip_runtime.h>

#define BB 4
#define NN 8192
#define MM 2048
#define CC 64
#define KK 16

#define TILES_PER_BATCH (MM / 16)      // 128 query tiles per batch
#define WAVES_PER_BLOCK 8
#define NSPLIT (NN / WAVES_PER_BLOCK)  // 1024 sources per wave
#define NTILES (NSPLIT / 16)           // 64 wmma tiles per wave

typedef __attribute__((ext_vector_type(2))) float v2f;
typedef __attribute__((ext_vector_type(8))) float v8f;

// Two-pointer merge of two sorted 16-lists, keep lowest 16.
__device__ __forceinline__ void merge16(const float* ka, const int* ja,
                                        const float* kb, const int* jb,
                                        float* ko, int* jo) {
  int ia = 0, ib = 0;
#pragma unroll
  for (int k = 0; k < 16; ++k) {
    float va = (ia < 16) ? ka[ia] : 3.4e38f;
    float vb = (ib < 16) ? kb[ib] : 3.4e38f;
    bool ta = va <= vb;
    ko[k] = ta ? va : vb;
    jo[k] = ta ? ja[ia] : jb[ib];
    ia += ta ? 1 : 0;
    ib += ta ? 0 : 1;
  }
}

__global__ __launch_bounds__(256) void knn_group_kernel(
    const float* __restrict__ pts,     // [B,3,N]
    const float* __restrict__ npts,    // [B,3,M]
    const float* __restrict__ feats,   // [B,C,N]
    float* __restrict__ out)           // [B,3+C,M,K]
{
  __shared__ float s_key[16][16][16];  // [query][list][k]
  __shared__ int   s_idx[16][16][16];
  __shared__ float s_key2[16][8][16];  // merge ping buffer
  __shared__ int   s_idx2[16][8][16];
  __shared__ int   s_fidx[16][16];     // final indices per query

  const int tid  = threadIdx.x;
  const int lane = tid & 31;
  const int w    = tid >> 5;     // wave 0..7
  const int half = lane >> 4;    // 0: lanes 0-15, 1: lanes 16-31
  const int j    = lane & 15;    // query column within the 16-query tile

  const int b  = blockIdx.x / TILES_PER_BATCH;
  const int m0 = (blockIdx.x % TILES_PER_BATCH) * 16;

  const float* px = pts + (size_t)(b * 3 + 0) * NN;
  const float* py = pts + (size_t)(b * 3 + 1) * NN;
  const float* pz = pts + (size_t)(b * 3 + 2) * NN;

  // ---- B operand: query columns, folded [-2qx,-2qy,-2qz,1] ----
  const int   m  = m0 + j;
  const float qx = npts[(size_t)(b * 3 + 0) * MM + m];
  const float qy = npts[(size_t)(b * 3 + 1) * MM + m];
  const float qz = npts[(size_t)(b * 3 + 2) * MM + m];
  v2f bq;
  bq.x = half ? (-2.0f * qz) : (-2.0f * qx);
  bq.y = half ? 1.0f         : (-2.0f * qy);

  // ---- per-lane sorted top-16 (ascending key) in registers ----
  float best[16]; int bidx[16];
#pragma unroll
  for (int k = 0; k < 16; ++k) { best[k] = 3.4e38f; bidx[k] = 0; }

  const int nbase  = w * NSPLIT;
  const int rowOff = half << 3;   // D rows this lane holds: rowOff..rowOff+7
  const int nsl    = j;           // source slot this lane supplies to A

  for (int tt = 0; tt < NTILES; ++tt) {
    const int n0 = nbase + tt * 16;
    const int ns = n0 + nsl;
    if (tt + 4 < NTILES) {
      __builtin_prefetch(&px[ns + 64], 0, 0);
      __builtin_prefetch(&py[ns + 64], 0, 0);
      __builtin_prefetch(&pz[ns + 64], 0, 0);
    }
    // A operand: source row [sx, sy, sz, ||s||^2]
    const float sx = px[ns], sy = py[ns], sz = pz[ns];
    const float nrm = fmaf(sx, sx, fmaf(sy, sy, sz * sz));
    v2f a;
    a.x = half ? sz  : sx;   // K = 0 (low half) / K = 2 (high half)
    a.y = half ? nrm : sy;   // K = 1 (low half) / K = 3 (high half)

    v8f zero = {};
    v8f acc = __builtin_amdgcn_wmma_f32_16x16x4_f32(
        false, a, false, bq, (short)0, zero, false, false);
    // acc[r] = ||s_{n0+rowOff+r}||^2 - 2 s.q_j  == dist - ||q_j||^2

#pragma unroll
    for (int r = 0; r < 8; ++r) {
      const float key = acc[r];
      if (key < best[15]) {
        best[15] = key;
        bidx[15] = n0 + rowOff + r;
        // one backward bubble pass restores sorted order (predicated swaps)
#pragma unroll
        for (int s = 15; s >= 1; --s) {
          const float k0 = best[s - 1], k1 = best[s];
          const int   i0 = bidx[s - 1], i1 = bidx[s];
          const bool  sw = k1 < k0;
          best[s - 1] = sw ? k1 : k0;  bidx[s - 1] = sw ? i1 : i0;
          best[s]     = sw ? k0 : k1;  bidx[s]     = sw ? i0 : i1;
        }
      }
    }
  }

  // ---- dump the 16 per-lane lists of each query to LDS ----
  const int listId = w * 2 + half;
#pragma unroll
  for (int k = 0; k < 16; ++k) {
    s_key[j][listId][k] = best[k];
    s_idx[j][listId][k] = bidx[k];
  }
  __syncthreads();

  // ---- log2 merge: 16 -> 8 -> 4 -> 2 -> 1 lists per query ----
  if (tid < 128) {                       // 16 queries x 8 merges
    int q = tid >> 3, p = tid & 7;
    merge16(&s_key[q][2 * p][0], &s_idx[q][2 * p][0],
            &s_key[q][2 * p + 1][0], &s_idx[q][2 * p + 1][0],
            &s_key2[q][p][0], &s_idx2[q][p][0]);
  }
  __syncthreads();
  if (tid < 64) {                        // 16 x 4
    int q = tid >> 2, p = tid & 3;
    merge16(&s_key2[q][2 * p][0], &s_idx2[q][2 * p][0],
            &s_key2[q][2 * p + 1][0], &s_idx2[q][2 * p + 1][0],
            &s_key[q][p][0], &s_idx[q][p][0]);
  }
  __syncthreads();
  if (tid < 32) {                        // 16 x 2
    int q = tid >> 1, p = tid & 1;
    merge16(&s_key[q][2 * p][0], &s_idx[q][2 * p][0],
            &s_key[q][2 * p + 1][0], &s_idx[q][2 * p + 1][0],
            &s_key2[q][p][0], &s_idx2[q][p][0]);
  }
  __syncthreads();
  if (tid < 16) {                        // final: keep indices only
    const float* ka = &s_key2[tid][0][0]; const int* ja = &s_idx2[tid][0][0];
    const float* kb = &s_key2[tid][1][0]; const int* jb = &s_idx2[tid][1][0];
    int ia = 0, ib = 0;
#pragma unroll
    for (int k = 0; k < 16; ++k) {
      float va = (ia < 16) ? ka[ia] : 3.4e38f;
      float vb = (ib < 16) ? kb[ib] : 3.4e38f;
      bool ta = va <= vb;
      s_fidx[tid][k] = ta ? ja[ia] : jb[ib];
      ia += ta ? 1 : 0;
      ib += ta ? 0 : 1;
    }
  }
  __syncthreads();

  // ---- cooperative gather + coalesced 64B-row writes ----
  // rows: (3+C) channels x 16 queries; each row = 16 consecutive output floats
  const float* fb = feats + (size_t)b * CC * NN;
  for (int r = tid; r < (3 + CC) * 16; r += 256) {
    const int ch = r >> 4;
    const int jq = r & 15;
    const int mq = m0 + jq;
    float vals[16];
    if (ch < 3) {
      const float  qc = npts[(size_t)(b * 3 + ch) * MM + mq];
      const float* pc = pts + (size_t)(b * 3 + ch) * NN;
#pragma unroll
      for (int k = 0; k < 16; ++k) vals[k] = pc[s_fidx[jq][k]] - qc;
    } else {
      const float* fc = fb + (size_t)(ch - 3) * NN;
#pragma unroll
      for (int k = 0; k < 16; ++k) vals[k] = fc[s_fidx[jq][k]];
    }
    float4* ov = reinterpret_cast<float4*>(
        out + ((size_t)(b * (3 + CC) + ch) * MM + mq) * 16);
    ov[0] = make_float4(vals[0],  vals[1],  vals[2],  vals[3]);
    ov[1] = make_float4(vals[4],  vals[5],  vals[6],  vals[7]);
    ov[2] = make_float4(vals[8],  vals[9],  vals[10], vals[11]);
    ov[3] = make_float4(vals[12], vals[13], vals[14], vals[15]);
  }
}

extern "C" void kernel_launch(void* const* d_in, const int* in_sizes, int n_in,
                              void* d_out, int out_size, void* d_ws, size_t ws_size,
                              hipStream_t stream) {
  (void)in_sizes; (void)n_in; (void)out_size; (void)d_ws; (void)ws_size;
  const float* pts   = (const float*)d_in[0];  // points   [B,3,N]
  const float* npts  = (const float*)d_in[1];  // new_points [B,3,M]
  const float* feats = (const float*)d_in[2];  // features [B,C,N]
  float* out = (float*)d_out;                  // [B,3+C,M,K]

  dim3 grid(BB * TILES_PER_BATCH);             // 512 blocks
  dim3 block(256);                             // 8 waves
  knn_group_kernel<<<grid, block, 0, stream>>>(pts, npts, feats, out);
}